// pocket_structure_net_82832739270715
// MI455X (gfx1250) — compile-verified
//
#include <hip/hip_runtime.h>
#include <hip/hip_bf16.h>

typedef __attribute__((ext_vector_type(2))) float v2f;
typedef __attribute__((ext_vector_type(8))) float v8f;

#define BS   32
#define L1   79
#define L2   192
#define F_IN 52
#define HID  128
#define E1N  160
#define E2N  400
#define SE_IN  15168      // L1*L2
#define SE_HID 2000
#define ND   (BS*L1)      // 2528 drug nodes
#define NP   (BS*L2)      // 6144 protein nodes
#define K1SPLIT 16        // SE1: 15168 = 16 * 948 (948 % 4 == 0)
#define K2SPLIT 4         // SE2: 2000  = 4  * 500 (500 % 4 == 0)

// ---------------------------------------------------------------------------
// f32 WMMA helper: D = A(16x4) * B(4x16) + C  (V_WMMA_F32_16X16X4_F32)
// ---------------------------------------------------------------------------
__device__ __forceinline__ v8f wmma_acc_f32(v2f a, v2f b, v8f c) {
  return __builtin_amdgcn_wmma_f32_16x16x4_f32(false, a, false, b, (short)0, c,
                                               false, false);
}

// ---------------------------------------------------------------------------
// Generic 16x16-tile GEMM: C[row0:+16, col0:+16] = act(A @ B + bias)
// One wave (32 lanes) per tile. A: MxK (lda), B: KxN (ldb), C (ldc).
// ---------------------------------------------------------------------------
__global__ __launch_bounds__(32)
void wmma_gemm_kernel(const float* __restrict__ A, const float* __restrict__ B,
                      const float* __restrict__ bias, float* __restrict__ C,
                      int K, int lda, int ldb, int ldc, int act) {
  const int row0 = blockIdx.x * 16;
  const int col0 = blockIdx.y * 16;
  const int lane = threadIdx.x & 31;
  const int lm   = lane & 15;
  const int hi   = lane >> 4;           // 0: K even pair, 1: K odd pair
  const float* Ar = A + (size_t)(row0 + lm) * lda;
  const float* Bc = B + col0 + lm;
  v8f acc = {};
  for (int k = 0; k < K; k += 4) {
    v2f a, b;
    a.x = Ar[k + 2 * hi];
    a.y = Ar[k + 2 * hi + 1];
    b.x = Bc[(size_t)(k + 2 * hi) * ldb];
    b.y = Bc[(size_t)(k + 2 * hi + 1) * ldb];
    acc = wmma_acc_f32(a, b, acc);
  }
  const float bv = bias ? bias[col0 + lm] : 0.0f;
  for (int r = 0; r < 8; ++r) {
    float v = acc[r] + bv;
    if (act) v = fmaxf(v, 0.0f);
    C[(size_t)(row0 + 8 * hi + r) * ldc + col0 + lm] = v;
  }
}

// ---------------------------------------------------------------------------
// SE GEMM with fixed M=32 (two 16-row A fragments share each B fragment) and
// K-split for occupancy; partials go to Cpart[kchunk][32][N], reduced later.
// ---------------------------------------------------------------------------
__global__ __launch_bounds__(32)
void wmma_se_kernel(const float* __restrict__ A, const float* __restrict__ B,
                    float* __restrict__ Cpart, int kchunk, int lda, int N) {
  const int col0 = blockIdx.x * 16;
  const int kc   = blockIdx.y;
  const int lane = threadIdx.x & 31;
  const int lm   = lane & 15;
  const int hi   = lane >> 4;
  const int kbeg = kc * kchunk;
  const float* A0 = A + (size_t)lm * lda;          // rows 0..15
  const float* A1 = A + (size_t)(16 + lm) * lda;   // rows 16..31
  const float* Bc = B + col0 + lm;
  v8f acc0 = {}, acc1 = {};
  for (int k = kbeg; k < kbeg + kchunk; k += 4) {
    v2f b, a0, a1;
    b.x  = Bc[(size_t)(k + 2 * hi) * N];
    b.y  = Bc[(size_t)(k + 2 * hi + 1) * N];
    a0.x = A0[k + 2 * hi];  a0.y = A0[k + 2 * hi + 1];
    a1.x = A1[k + 2 * hi];  a1.y = A1[k + 2 * hi + 1];
    acc0 = wmma_acc_f32(a0, b, acc0);
    acc1 = wmma_acc_f32(a1, b, acc1);
  }
  float* Cp = Cpart + (size_t)kc * 32 * N;
  for (int r = 0; r < 8; ++r) {
    Cp[(size_t)(8 * hi + r) * N + col0 + lm]      = acc0[r];
    Cp[(size_t)(16 + 8 * hi + r) * N + col0 + lm] = acc1[r];
  }
}

// ---------------------------------------------------------------------------
// Per-node attention coefficients: hs[n] = h[n,:]·a_s, hd[n] = h[n,:]·a_d
// ---------------------------------------------------------------------------
__global__ __launch_bounds__(128)
void gat_coef_kernel(const float* __restrict__ h, const float* __restrict__ a_s,
                     const float* __restrict__ a_d, float* __restrict__ hs,
                     float* __restrict__ hd) {
  __shared__ float rs[128], rd[128];
  const int n = blockIdx.x;
  const int t = threadIdx.x;
  const float hv = h[(size_t)n * HID + t];
  rs[t] = hv * a_s[t];
  rd[t] = hv * a_d[t];
  __syncthreads();
  for (int s = 64; s > 0; s >>= 1) {
    if (t < s) { rs[t] += rs[t + s]; rd[t] += rd[t + s]; }
    __syncthreads();
  }
  if (t == 0) { hs[n] = rs[0]; hd[n] = rd[0]; }
}

__device__ __forceinline__ float lrelu02(float x) {
  return (x > 0.0f) ? x : 0.2f * x;
}

// ---------------------------------------------------------------------------
// GAT segment-softmax aggregation (gather form, deterministic, no atomics).
// One block per destination node; edges of graph g are the contiguous slice
// [g*e_per, (g+1)*e_per). Self-loop handled explicitly. Writes relu(gat+bias)
// to `out` and adds into the residual accumulator `acc`.
// ---------------------------------------------------------------------------
__global__ __launch_bounds__(128)
void gat_aggregate_kernel(const float* __restrict__ h, const float* __restrict__ hs,
                          const float* __restrict__ hd, const int* __restrict__ src,
                          const int* __restrict__ dst, int e_per, int n_per,
                          const float* __restrict__ bias, float* __restrict__ out,
                          float* __restrict__ acc, int accumulate) {
  __shared__ float red[128];
  __shared__ float s_m, s_denom;
  const int d = blockIdx.x;
  const int t = threadIdx.x;
  const int base = (d / n_per) * e_per;
  const float hdd = hd[d];

  // pass 1: segment max (incoming edges + self loop)
  float local = -3.402823466e+38f;
  for (int k = t; k < e_per; k += 128)
    if (dst[base + k] == d)
      local = fmaxf(local, lrelu02(hs[src[base + k]] + hdd));
  if (t == 0) local = fmaxf(local, lrelu02(hs[d] + hdd));
  red[t] = local; __syncthreads();
  for (int s = 64; s > 0; s >>= 1) { if (t < s) red[t] = fmaxf(red[t], red[t + s]); __syncthreads(); }
  if (t == 0) s_m = red[0];
  __syncthreads();
  const float m = s_m;

  // pass 2: softmax denominator
  local = 0.0f;
  for (int k = t; k < e_per; k += 128)
    if (dst[base + k] == d)
      local += __expf(lrelu02(hs[src[base + k]] + hdd) - m);
  if (t == 0) local += __expf(lrelu02(hs[d] + hdd) - m);
  red[t] = local; __syncthreads();
  for (int s = 64; s > 0; s >>= 1) { if (t < s) red[t] += red[t + s]; __syncthreads(); }
  if (t == 0) s_denom = red[0];
  __syncthreads();
  const float inv_denom = 1.0f / s_denom;

  // pass 3: thread t owns hidden dim t; serial edge scan (deterministic)
  float av = 0.0f;
  for (int k = 0; k < e_per; ++k) {
    if (dst[base + k] == d) {
      const int sk = src[base + k];
      const float w = __expf(lrelu02(hs[sk] + hdd) - m) * inv_denom;
      av += w * h[(size_t)sk * HID + t];
    }
  }
  { // self loop
    const float w = __expf(lrelu02(hs[d] + hdd) - m) * inv_denom;
    av += w * h[(size_t)d * HID + t];
  }
  const float v = fmaxf(av + bias[t], 0.0f);   // relu(GAT out)
  out[(size_t)d * HID + t] = v;
  if (accumulate) acc[(size_t)d * HID + t] += v;
  else            acc[(size_t)d * HID + t]  = v;
}

// ---------------------------------------------------------------------------
// Small element-wise / reduction kernels
// ---------------------------------------------------------------------------
__global__ void rowsum_kernel(const float* __restrict__ x, float* __restrict__ s, int n) {
  const int i = blockIdx.x * blockDim.x + threadIdx.x;
  if (i >= n) return;
  float a = 0.0f;
  for (int k = 0; k < HID; ++k) a += x[(size_t)i * HID + k];
  s[i] = a;
}

__global__ void cin_kernel(const float* __restrict__ sl, const float* __restrict__ sp,
                           float* __restrict__ cin) {
  const int idx = blockIdx.x * blockDim.x + threadIdx.x;
  if (idx >= BS * SE_IN) return;
  const int b = idx / SE_IN;
  const int ij = idx - b * SE_IN;
  const int i = ij / L2;
  const int j = ij - i * L2;
  cin[idx] = (sl[b * L1 + i] + sp[b * L2 + j]) * (1.0f / 256.0f);
}

__global__ void c1_reduce_kernel(const float* __restrict__ part,
                                 const float* __restrict__ bias, float* __restrict__ c1) {
  const int idx = blockIdx.x * blockDim.x + threadIdx.x;
  if (idx >= 32 * SE_HID) return;
  float s = bias[idx % SE_HID];
  for (int c = 0; c < K1SPLIT; ++c) s += part[(size_t)c * 32 * SE_HID + idx];
  c1[idx] = fmaxf(s, 0.0f);
}

__global__ void wbuf_kernel(const float* __restrict__ part, const float* __restrict__ bias,
                            const float* __restrict__ inter, float* __restrict__ w) {
  const int idx = blockIdx.x * blockDim.x + threadIdx.x;
  if (idx >= BS * SE_IN) return;
  float s = bias[idx % SE_IN];
  for (int c = 0; c < K2SPLIT; ++c) s += part[(size_t)c * BS * SE_IN + idx];
  const float sig = 1.0f / (1.0f + __expf(-s));
  w[idx] = sig / inter[idx];   // fold inter_att = 1/inter into the weight
}

__global__ void rl_kernel(const float* __restrict__ w, float* __restrict__ rl) {
  const int idx = blockIdx.x * blockDim.x + threadIdx.x;
  if (idx >= BS * L1) return;
  const int b = idx / L1, i = idx % L1;
  const float* wr = w + (size_t)b * SE_IN + (size_t)i * L2;
  float s = 0.0f;
  for (int j = 0; j < L2; ++j) s += wr[j];
  rl[idx] = s;
}

__global__ void rp_kernel(const float* __restrict__ w, float* __restrict__ rp) {
  const int idx = blockIdx.x * blockDim.x + threadIdx.x;
  if (idx >= BS * L2) return;
  const int b = idx / L2, j = idx % L2;
  const float* wb = w + (size_t)b * SE_IN + j;
  float s = 0.0f;
  for (int i = 0; i < L1; ++i) s += wb[(size_t)i * L2];
  rp[idx] = s;
}

// xx[b,h] = (1/SE_IN) * (h<128 ? sum_i xl[b,i,h]*rl[b,i] : sum_j xp[b,j,h-128]*rp[b,j])
__global__ void xx_kernel(const float* __restrict__ xl, const float* __restrict__ xp,
                          const float* __restrict__ rl, const float* __restrict__ rp,
                          float* __restrict__ xx) {
  const int idx = blockIdx.x * blockDim.x + threadIdx.x;
  if (idx >= BS * 256) return;
  const int b = idx / 256, hh = idx % 256;
  float s = 0.0f;
  if (hh < HID) {
    for (int i = 0; i < L1; ++i) s += xl[(size_t)(b * L1 + i) * HID + hh] * rl[b * L1 + i];
  } else {
    const int h2 = hh - HID;
    for (int j = 0; j < L2; ++j) s += xp[(size_t)(b * L2 + j) * HID + h2] * rp[b * L2 + j];
  }
  xx[idx] = s * (1.0f / (float)SE_IN);
}

__global__ void dense_kernel(const float* __restrict__ X, const float* __restrict__ W,
                             const float* __restrict__ B, float* __restrict__ Y,
                             int M, int K, int N, int act) {
  const int idx = blockIdx.x * blockDim.x + threadIdx.x;
  if (idx >= M * N) return;
  const int m = idx / N, n = idx % N;
  float s = B[n];
  for (int k = 0; k < K; ++k) s += X[(size_t)m * K + k] * W[(size_t)k * N + n];
  if (act) s = fmaxf(s, 0.0f);
  Y[idx] = s;
}

__global__ void relu_copy_kernel(const float* __restrict__ x, float* __restrict__ y, int n) {
  const int i = blockIdx.x * blockDim.x + threadIdx.x;
  if (i < n) y[i] = fmaxf(x[i], 0.0f);
}

// ---------------------------------------------------------------------------
// Host launcher
// ---------------------------------------------------------------------------
static inline float* wsf(void* base, size_t& off, size_t count) {
  float* p = (float*)((char*)base + off);
  off = (off + count * sizeof(float) + 255) & ~(size_t)255;
  return p;
}

extern "C" void kernel_launch(void* const* d_in, const int* in_sizes, int n_in,
                              void* d_out, int out_size, void* d_ws, size_t ws_size,
                              hipStream_t stream) {
  (void)in_sizes; (void)n_in; (void)out_size; (void)ws_size;
  const float* x1     = (const float*)d_in[0];
  const float* x2     = (const float*)d_in[1];
  const float* inter  = (const float*)d_in[2];
  const int*   dei    = (const int*)  d_in[3];   // (2, BS*E1N)
  const int*   pei    = (const int*)  d_in[4];   // (2, BS*E2N)
  const float* w1     = (const float*)d_in[5];
  const float* b1     = (const float*)d_in[6];
  const float* w2     = (const float*)d_in[7];
  const float* b2     = (const float*)d_in[8];
  const float* gat_w  = (const float*)d_in[9];   // (6,128,128)
  const float* gat_as = (const float*)d_in[10];  // (6,128)
  const float* gat_ad = (const float*)d_in[11];
  const float* gat_b  = (const float*)d_in[12];
  const float* se1_w  = (const float*)d_in[13];
  const float* se1_b  = (const float*)d_in[14];
  const float* se2_w  = (const float*)d_in[15];
  const float* se2_b  = (const float*)d_in[16];
  const float* fc1_w  = (const float*)d_in[17];
  const float* fc1_b  = (const float*)d_in[18];
  const float* fc2_w  = (const float*)d_in[19];
  const float* fc2_b  = (const float*)d_in[20];
  const float* fc3_w  = (const float*)d_in[21];
  const float* fc3_b  = (const float*)d_in[22];
  const float* fc4_w  = (const float*)d_in[23];
  const float* fc4_b  = (const float*)d_in[24];
  const float* out_w  = (const float*)d_in[25];
  const float* out_b  = (const float*)d_in[26];
  float* out = (float*)d_out;                    // [32] out, then [32*512] h1

  const int* dsrc = dei;                const int* ddst = dei + BS * E1N;
  const int* psrc = pei;                const int* pdst = pei + BS * E2N;

  size_t off = 0;
  float* e1     = wsf(d_ws, off, (size_t)ND * HID);
  float* e2     = wsf(d_ws, off, (size_t)NP * HID);
  float* t1     = wsf(d_ws, off, (size_t)NP * HID);   // MLP hidden (reused)
  float* hbuf   = wsf(d_ws, off, (size_t)NP * HID);   // GAT transform (reused)
  float* hs     = wsf(d_ws, off, (size_t)NP);
  float* hd     = wsf(d_ws, off, (size_t)NP);
  float* dA     = wsf(d_ws, off, (size_t)ND * HID);
  float* dB     = wsf(d_ws, off, (size_t)ND * HID);
  float* pA     = wsf(d_ws, off, (size_t)NP * HID);
  float* pB     = wsf(d_ws, off, (size_t)NP * HID);
  float* xl_acc = wsf(d_ws, off, (size_t)ND * HID);
  float* xp_acc = wsf(d_ws, off, (size_t)NP * HID);
  float* sl     = wsf(d_ws, off, (size_t)ND);
  float* sp     = wsf(d_ws, off, (size_t)NP);
  float* cin    = wsf(d_ws, off, (size_t)BS * SE_IN);
  float* c1part = wsf(d_ws, off, (size_t)K1SPLIT * 32 * SE_HID);
  float* c1     = wsf(d_ws, off, (size_t)32 * SE_HID);
  float* c2part = wsf(d_ws, off, (size_t)K2SPLIT * BS * SE_IN);
  float* wbuf   = wsf(d_ws, off, (size_t)BS * SE_IN);
  float* rl     = wsf(d_ws, off, (size_t)BS * L1);
  float* rp     = wsf(d_ws, off, (size_t)BS * L2);
  float* xx     = wsf(d_ws, off, (size_t)BS * 256);
  float* a1     = wsf(d_ws, off, (size_t)BS * 512);
  float* a2     = wsf(d_ws, off, (size_t)BS * 256);
  float* a3     = wsf(d_ws, off, (size_t)BS * 128);
  float* a4     = wsf(d_ws, off, (size_t)BS * 64);

  // ---- shared node-embedding MLP (WMMA) ----
  wmma_gemm_kernel<<<dim3(ND / 16, HID / 16), 32, 0, stream>>>(x1, w1, b1, t1, F_IN, F_IN, HID, HID, 1);
  wmma_gemm_kernel<<<dim3(ND / 16, HID / 16), 32, 0, stream>>>(t1, w2, b2, e1, HID, HID, HID, HID, 1);
  wmma_gemm_kernel<<<dim3(NP / 16, HID / 16), 32, 0, stream>>>(x2, w1, b1, t1, F_IN, F_IN, HID, HID, 1);
  wmma_gemm_kernel<<<dim3(NP / 16, HID / 16), 32, 0, stream>>>(t1, w2, b2, e2, HID, HID, HID, HID, 1);

  // ---- drug branch: 3 GAT layers (layers 0..2) ----
  {
    const float* cur = e1;
    float* outs[2] = { dA, dB };
    for (int l = 0; l < 3; ++l) {
      wmma_gemm_kernel<<<dim3(ND / 16, HID / 16), 32, 0, stream>>>(
          cur, gat_w + (size_t)l * HID * HID, nullptr, hbuf, HID, HID, HID, HID, 0);
      gat_coef_kernel<<<ND, 128, 0, stream>>>(hbuf, gat_as + l * HID, gat_ad + l * HID, hs, hd);
      gat_aggregate_kernel<<<ND, 128, 0, stream>>>(hbuf, hs, hd, dsrc, ddst, E1N, L1,
                                                   gat_b + l * HID, outs[l & 1], xl_acc, l > 0);
      cur = outs[l & 1];
    }
  }
  // ---- protein branch: 3 GAT layers (layers 3..5) ----
  {
    const float* cur = e2;
    float* outs[2] = { pA, pB };
    for (int l = 0; l < 3; ++l) {
      const int gl = l + 3;
      wmma_gemm_kernel<<<dim3(NP / 16, HID / 16), 32, 0, stream>>>(
          cur, gat_w + (size_t)gl * HID * HID, nullptr, hbuf, HID, HID, HID, HID, 0);
      gat_coef_kernel<<<NP, 128, 0, stream>>>(hbuf, gat_as + gl * HID, gat_ad + gl * HID, hs, hd);
      gat_aggregate_kernel<<<NP, 128, 0, stream>>>(hbuf, hs, hd, psrc, pdst, E2N, L2,
                                                   gat_b + gl * HID, outs[l & 1], xp_acc, l > 0);
      cur = outs[l & 1];
    }
  }

  // ---- SE block input: mean over concat dims == (rowsum_l + rowsum_p)/256 ----
  rowsum_kernel<<<(ND + 127) / 128, 128, 0, stream>>>(xl_acc, sl, ND);
  rowsum_kernel<<<(NP + 127) / 128, 128, 0, stream>>>(xp_acc, sp, NP);
  cin_kernel<<<(BS * SE_IN + 255) / 256, 256, 0, stream>>>(sl, sp, cin);

  // ---- SE GEMM 1: (32,15168)@(15168,2000), K-split 16 (bandwidth-dominant) ----
  wmma_se_kernel<<<dim3(SE_HID / 16, K1SPLIT), 32, 0, stream>>>(cin, se1_w, c1part,
                                                               SE_IN / K1SPLIT, SE_IN, SE_HID);
  c1_reduce_kernel<<<(32 * SE_HID + 255) / 256, 256, 0, stream>>>(c1part, se1_b, c1);

  // ---- SE GEMM 2: (32,2000)@(2000,15168), K-split 4 ----
  wmma_se_kernel<<<dim3(SE_IN / 16, K2SPLIT), 32, 0, stream>>>(c1, se2_w, c2part,
                                                               SE_HID / K2SPLIT, SE_HID, SE_IN);
  wbuf_kernel<<<(BS * SE_IN + 255) / 256, 256, 0, stream>>>(c2part, se2_b, inter, wbuf);

  // ---- factorized fusion reduce ----
  rl_kernel<<<(BS * L1 + 127) / 128, 128, 0, stream>>>(wbuf, rl);
  rp_kernel<<<(BS * L2 + 127) / 128, 128, 0, stream>>>(wbuf, rp);
  xx_kernel<<<(BS * 256 + 127) / 128, 128, 0, stream>>>(xl_acc, xp_acc, rl, rp, xx);

  // ---- head MLP; h1 (pre-relu) is the second output ----
  float* h1 = out + 32;
  dense_kernel<<<(BS * 512 + 127) / 128, 128, 0, stream>>>(xx, fc1_w, fc1_b, h1, BS, 256, 512, 0);
  relu_copy_kernel<<<(BS * 512 + 127) / 128, 128, 0, stream>>>(h1, a1, BS * 512);
  dense_kernel<<<(BS * 256 + 127) / 128, 128, 0, stream>>>(a1, fc2_w, fc2_b, a2, BS, 512, 256, 1);
  dense_kernel<<<(BS * 128 + 127) / 128, 128, 0, stream>>>(a2, fc3_w, fc3_b, a3, BS, 256, 128, 1);
  dense_kernel<<<(BS * 64 + 127) / 128, 128, 0, stream>>>(a3, fc4_w, fc4_b, a4, BS, 128, 64, 1);
  dense_kernel<<<(BS * 1 + 127) / 128, 128, 0, stream>>>(a4, out_w, out_b, out, BS, 64, 1, 0);
}